// STOI_79860621902377
// MI455X (gfx1250) — compile-verified
//
#include <hip/hip_runtime.h>
#include <math.h>

// ---------------------------------------------------------------------------
// STOI pipeline for gfx1250 (MI455X).  f32 WMMA (V_WMMA_F32_16X16X4_F32) does
// both the DFT GEMM and the third-octave band GEMM.  The Hann window is folded
// into a pre-packed, lane-ordered DFT-fragment stream (one coalesced b128 load
// per k-step), and bin-groups with no OBM membership are skipped exactly.
// Data-dependent sizes (K, T, J) live in a device meta block; kernels use
// worst-case grids with device-side guards -> graph-capture safe.
// ---------------------------------------------------------------------------

typedef float v2f __attribute__((ext_vector_type(2)));
typedef float v8f __attribute__((ext_vector_type(8)));

#define SIG_LEN   8388608
#define NFRAME    256
#define HOP       128
#define NF        65534      // silence-detection frames
#define NF_PAD    65536
#define NBIN      257
#define NGROUP    17         // 17 groups of 16 bins (bins >= 257 zero)
#define NUMBAND   15
#define NCTX      30
#define EPSF      1.1920928955078125e-07f
#define CLIPF     6.6234132519034908f   // 1 + 10^(15/20)
#define XS_ALLOC  8392704    // padded resynth buffer (covers max tile span)
#define SEG_BLOCKS 4096
#define NFRAG     (NGROUP * 64 * 32)    // float4 per (group,kstep,lane) = 34816
#define NOBMF     (NGROUP * 4 * 32)     // float2 per (group,qstep,lane) = 2176
#define TWO_PI    6.283185307179586

static __device__ __forceinline__ v8f wmma4(v2f a, v2f b, v8f c) {
  // D(16x16,f32) = A(16x4,f32) x B(4x16,f32) + C
  return __builtin_amdgcn_wmma_f32_16x16x4_f32(false, a, false, b,
                                               (short)0, c, false, false);
}

// --------------------------- table initialization ---------------------------
// frag[((g*64+kk)*32+lane)] = float4{ w[n0]*cos(th0), w[n1]*cos(th1),
//                                     w[n0]*sin(th0), w[n1]*sin(th1) }
// with n0 = 4*kk + 2*kh, kb = 16*g + (lane&15), zero past bin 256.
__global__ void init_tables_kernel(const float* __restrict__ obm,
                                   float* __restrict__ w,
                                   float4* __restrict__ frag,
                                   float2* __restrict__ obmf,
                                   int* __restrict__ gmask) {
  int idx = blockIdx.x * 256 + threadIdx.x;
  if (idx < NFRAME) {
    // np.hanning(258)[1:-1]
    double th = TWO_PI * (double)(idx + 1) / 257.0;
    w[idx] = (float)(0.5 * (1.0 - cos(th)));
  }
  if (idx < NFRAG) {
    int lane = idx & 31;
    int kk   = (idx >> 5) & 63;
    int g    = idx >> 11;
    int kb   = g * 16 + (lane & 15);
    int n0   = kk * 4 + ((lane >> 4) & 1) * 2;
    double w0 = 0.5 * (1.0 - cos(TWO_PI * (double)(n0 + 1) / 257.0));
    double w1 = 0.5 * (1.0 - cos(TWO_PI * (double)(n0 + 2) / 257.0));
    float4 f = make_float4(0.f, 0.f, 0.f, 0.f);
    if (kb < NBIN) {
      double t0 = TWO_PI * (double)kb * (double)n0 / 512.0;
      double t1 = TWO_PI * (double)kb * (double)(n0 + 1) / 512.0;
      f = make_float4((float)(w0 * cos(t0)), (float)(w1 * cos(t1)),
                      (float)(w0 * sin(t0)), (float)(w1 * sin(t1)));
    }
    frag[idx] = f;
  }
  if (idx < NOBMF) {
    int lane = idx & 31;
    int q    = (idx >> 5) & 3;
    int g    = idx >> 7;
    int band = lane & 15;
    int bin0 = g * 16 + q * 4 + ((lane >> 4) & 1) * 2;
    float e0 = 0.f, e1 = 0.f;
    if (band < NUMBAND) {
      if (bin0 < NBIN)     e0 = obm[band * NBIN + bin0];
      if (bin0 + 1 < NBIN) e1 = obm[band * NBIN + bin0 + 1];
    }
    obmf[idx] = make_float2(e0, e1);
  }
  if (idx < NGROUP) {
    int any = 0;
    for (int band = 0; band < NUMBAND; ++band)
      for (int b = idx * 16; b < idx * 16 + 16 && b < NBIN; ++b)
        any |= (obm[band * NBIN + b] != 0.f);
    gmask[idx] = any;
  }
}

// ------------------------------ frame energies ------------------------------
__global__ void frame_energy_kernel(const float* __restrict__ ref,
                                    const float* __restrict__ w,
                                    float* __restrict__ energy) {
  int lane = threadIdx.x & 31, wv = threadIdx.x >> 5;
  int frame = blockIdx.x * 8 + wv;           // grid 8192 x 256 -> 65536 frames
  float ss = 0.f;
  if (frame < NF) {
    size_t base = (size_t)frame * HOP;
    for (int s = lane; s < NFRAME; s += 32) {
      float v = ref[base + s] * w[s];
      ss += v * v;
    }
  }
  for (int d = 16; d > 0; d >>= 1) ss += __shfl_xor(ss, d, 32);
  if (lane == 0)
    energy[frame] = (frame < NF) ? 20.f * log10f(sqrtf(ss) + EPSF) : -1e30f;
}

__global__ void max_part_kernel(const float* __restrict__ energy,
                                float* __restrict__ part) {
  __shared__ float red[256];
  int tid = threadIdx.x;
  float m = -1e30f;
  for (int i = blockIdx.x * 256 + tid; i < NF_PAD; i += 256 * 256)
    m = fmaxf(m, energy[i]);
  red[tid] = m; __syncthreads();
  for (int s = 128; s > 0; s >>= 1) {
    if (tid < s) red[tid] = fmaxf(red[tid], red[tid + s]);
    __syncthreads();
  }
  if (tid == 0) part[blockIdx.x] = red[0];
}

__global__ void max_final_kernel(const float* __restrict__ part,
                                 float* __restrict__ metaf) {
  __shared__ float red[256];
  int tid = threadIdx.x;
  red[tid] = part[tid]; __syncthreads();
  for (int s = 128; s > 0; s >>= 1) {
    if (tid < s) red[tid] = fmaxf(red[tid], red[tid + s]);
    __syncthreads();
  }
  if (tid == 0) metaf[0] = red[0];
}

// ------------------ ordered compaction of non-silent frames -----------------
__global__ void scan_kernel(const float* __restrict__ energy,
                            const float* __restrict__ metaf,
                            int* __restrict__ kept,
                            int* __restrict__ metai) {
  __shared__ int sc[1024];
  int tid = threadIdx.x;
  float thresh = metaf[0] - 40.0f;
  int offset = 0;
  for (int chunk = 0; chunk < NF_PAD; chunk += 1024) {
    int i = chunk + tid;
    int flag = (i < NF) && (energy[i] > thresh);
    sc[tid] = flag;
    __syncthreads();
    for (int d = 1; d < 1024; d <<= 1) {        // Hillis-Steele inclusive scan
      int v = (tid >= d) ? sc[tid - d] : 0;
      __syncthreads();
      sc[tid] += v;
      __syncthreads();
    }
    if (flag) kept[offset + sc[tid] - 1] = i;
    int total = sc[1023];
    __syncthreads();
    offset += total;
  }
  if (tid == 0) {
    int K = offset;
    metai[0] = K;                 // kept frames
    metai[1] = K - 1;             // T (STFT frames)
    metai[2] = K - 1 - (NCTX - 1);// J (segments)
  }
}

// --------------------- overlap-add resynthesis (gather) ---------------------
__global__ void overlap_add_kernel(const float* __restrict__ ref,
                                   const float* __restrict__ inf,
                                   const float* __restrict__ w,
                                   const int* __restrict__ kept,
                                   const int* __restrict__ metai,
                                   float* __restrict__ xs,
                                   float* __restrict__ ys) {
  size_t p = (size_t)blockIdx.x * 256 + threadIdx.x;   // grid covers XS_ALLOC
  int K = metai[0];
  size_t end = (size_t)K * HOP + HOP;                  // (K-1)*hop + framelen
  float xv = 0.f, yv = 0.f;
  if (p < end) {
    int j1 = (int)(p >> 7), o1 = (int)(p & 127);
    if (j1 < K) {
      size_t s = (size_t)kept[j1] * HOP + o1;
      xv += ref[s] * w[o1]; yv += inf[s] * w[o1];
    }
    int j0 = j1 - 1, o0 = o1 + 128;
    if (j0 >= 0) {
      size_t s = (size_t)kept[j0] * HOP + o0;
      xv += ref[s] * w[o0]; yv += inf[s] * w[o0];
    }
  }
  xs[p] = xv; ys[p] = yv;   // zero-fill padding -> later tile loads are clean
}

// ------------- WMMA workhorse: DFT + |.|^2 + band GEMM, fused ---------------
// 2 waves per block; each wave owns a 16-frame tile (frames overlap by 128, so
// the tile's raw 2176-sample span is staged in LDS; the window lives in the
// packed B-fragment stream, so A operands are raw LDS samples).
__global__ __launch_bounds__(64) void dft_band_kernel(
    const float* __restrict__ xs, const float* __restrict__ ys,
    const float4* __restrict__ frag, const float2* __restrict__ obmf,
    const int* __restrict__ gmask, const int* __restrict__ metai,
    float* __restrict__ xtob, float* __restrict__ ytob) {
  __shared__ float sxl[2][2176];
  __shared__ float syl[2][2176];
  __shared__ float sp[2][2][256];      // power tile staging (C/D -> A layout)

  const int tid = threadIdx.x, lane = tid & 31, wv = tid >> 5;
  const int T = metai[1];
  if (blockIdx.x * 32 >= T) return;    // block-uniform: whole block past T

  const int tile = blockIdx.x * 2 + wv;          // grid 2048 -> 4096 tiles
  const int t0 = tile * 16;
  const size_t base = (size_t)t0 * HOP;
  for (int i = lane; i < 2176; i += 32) {
    sxl[wv][i] = xs[base + i];
    syl[wv][i] = ys[base + i];
  }
  __syncthreads();

  const int m = lane & 15;             // A-row / B-col / D-col index
  const int kh = (lane >> 4) & 1;      // k-half select for A/B fragments
  v8f cbx = {}; v8f cby = {};          // band-energy^2 accumulators (16x16)

  for (int g = 0; g < NGROUP; ++g) {
    if (!gmask[g]) continue;           // group has no band membership: exact 0
    if (g + 1 < NGROUP)                // warm next group's fragment stream
      __builtin_prefetch((const void*)(frag + (size_t)(g + 1) * 2048 + lane), 0, 0);

    v8f cxr = {}, cxi = {}, cyr = {}, cyi = {};
    const float4* fp = frag + (size_t)g * 2048 + lane;
#pragma unroll 4
    for (int kk = 0; kk < 64; ++kk) {
      const float4 f = fp[kk * 32];    // one coalesced b128: {bc0,bc1,bs0,bs1}
      const int n = kk * 4 + kh * 2;   // within-frame sample index
      v2f ax, ay, bc, bs;
      ax[0] = sxl[wv][m * 128 + n];  ax[1] = sxl[wv][m * 128 + n + 1];
      ay[0] = syl[wv][m * 128 + n];  ay[1] = syl[wv][m * 128 + n + 1];
      bc[0] = f.x; bc[1] = f.y;
      bs[0] = f.z; bs[1] = f.w;
      cxr = wmma4(ax, bc, cxr);
      cxi = wmma4(ax, bs, cxi);
      cyr = wmma4(ay, bc, cyr);
      cyi = wmma4(ay, bs, cyi);
    }
    __syncthreads();                   // previous group's sp reads are done
#pragma unroll
    for (int r = 0; r < 8; ++r) {      // |X|^2 tile, C/D layout -> LDS
      sp[wv][0][(r + 8 * kh) * 16 + m] = cxr[r] * cxr[r] + cxi[r] * cxi[r];
      sp[wv][1][(r + 8 * kh) * 16 + m] = cyr[r] * cyr[r] + cyi[r] * cyi[r];
    }
    __syncthreads();
    const float2* op = obmf + (size_t)g * 128 + lane;
#pragma unroll
    for (int q = 0; q < 4; ++q) {      // band GEMM over this group's 16 bins
      const int kq = q * 4 + kh * 2;
      const float2 ob = op[q * 32];    // one coalesced b64
      v2f apx, apy, bo;
      apx[0] = sp[wv][0][m * 16 + kq]; apx[1] = sp[wv][0][m * 16 + kq + 1];
      apy[0] = sp[wv][1][m * 16 + kq]; apy[1] = sp[wv][1][m * 16 + kq + 1];
      bo[0] = ob.x; bo[1] = ob.y;
      cbx = wmma4(apx, bo, cbx);
      cby = wmma4(apy, bo, cby);
    }
  }
#pragma unroll
  for (int r = 0; r < 8; ++r) {
    int t = t0 + r + 8 * kh;
    if (t < T) {
      xtob[(size_t)t * 16 + m] = sqrtf(fmaxf(cbx[r], 0.f));
      ytob[(size_t)t * 16 + m] = sqrtf(fmaxf(cby[r], 0.f));
    }
  }
}

// --------------- per-segment clipped normalized correlations ----------------
__global__ void seg_corr_kernel(const float* __restrict__ xtob,
                                const float* __restrict__ ytob,
                                const int* __restrict__ metai,
                                float* __restrict__ part) {
  __shared__ float red[256];
  int tid = threadIdx.x;
  long gid = (long)blockIdx.x * 256 + tid;
  int j = (int)(gid >> 4);
  int band = (int)(gid & 15);
  int J = metai[2];
  float contrib = 0.f;
  if (band < NUMBAND && j < J) {
    float x[NCTX], y[NCTX];
    float sx2 = 0.f, sy2 = 0.f;
    for (int n = 0; n < NCTX; ++n) {
      float xv = xtob[(size_t)(j + n) * 16 + band];
      float yv = ytob[(size_t)(j + n) * 16 + band];
      x[n] = xv; y[n] = yv; sx2 += xv * xv; sy2 += yv * yv;
    }
    float alpha = sqrtf(sx2) / (sqrtf(sy2) + EPSF);
    float sy = 0.f, sx = 0.f;
    for (int n = 0; n < NCTX; ++n) {
      float yp = fminf(y[n] * alpha, x[n] * CLIPF);
      y[n] = yp; sy += yp; sx += x[n];
    }
    float my = sy * (1.f / NCTX), mx = sx * (1.f / NCTX);
    float syy = 0.f, sxx = 0.f, sxy = 0.f;
    for (int n = 0; n < NCTX; ++n) {
      float dy = y[n] - my, dx = x[n] - mx;
      syy += dy * dy; sxx += dx * dx; sxy += dy * dx;
    }
    contrib = sxy / ((sqrtf(syy) + EPSF) * (sqrtf(sxx) + EPSF));
  }
  red[tid] = contrib; __syncthreads();
  for (int s = 128; s > 0; s >>= 1) {
    if (tid < s) red[tid] += red[tid + s];
    __syncthreads();
  }
  if (tid == 0) part[blockIdx.x] = red[0];
}

__global__ void finalize_kernel(const float* __restrict__ part,
                                const int* __restrict__ metai,
                                float* __restrict__ out) {
  __shared__ float red[256];
  int tid = threadIdx.x;
  float s = 0.f;
  for (int i = tid; i < SEG_BLOCKS; i += 256) s += part[i];
  red[tid] = s; __syncthreads();
  for (int st = 128; st > 0; st >>= 1) {
    if (tid < st) red[tid] += red[tid + st];
    __syncthreads();
  }
  if (tid == 0) {
    int T = metai[1], J = metai[2];
    out[0] = (T < NCTX) ? 1e-5f : red[0] / ((float)J * (float)NUMBAND);
  }
}

// --------------------------------- launcher ---------------------------------
extern "C" void kernel_launch(void* const* d_in, const int* in_sizes, int n_in,
                              void* d_out, int out_size, void* d_ws, size_t ws_size,
                              hipStream_t stream) {
  (void)in_sizes; (void)n_in; (void)out_size; (void)ws_size;
  const float* ref = (const float*)d_in[0];
  const float* inf = (const float*)d_in[1];
  const float* obm = (const float*)d_in[2];    // (15, 257) row-major
  float* out = (float*)d_out;
  char* ws = (char*)d_ws;

  size_t off = 0;
  auto grab = [&](size_t bytes) {
    size_t o = off;
    off += (bytes + 255) & ~(size_t)255;
    return o;
  };
  float*  w      = (float*) (ws + grab((size_t)NFRAME * 4));
  float4* frag   = (float4*)(ws + grab((size_t)NFRAG * 16));
  float2* obmf   = (float2*)(ws + grab((size_t)NOBMF * 8));
  int*    gmask  = (int*)   (ws + grab((size_t)NGROUP * 4));
  float*  energy = (float*) (ws + grab((size_t)NF_PAD * 4));
  float*  partmx = (float*) (ws + grab((size_t)256 * 4));
  float*  metaf  = (float*) (ws + grab(64));
  int*    metai  = (int*)   (ws + grab(64));
  int*    kept   = (int*)   (ws + grab((size_t)NF_PAD * 4));
  float*  xsil   = (float*) (ws + grab((size_t)XS_ALLOC * 4));
  float*  ysil   = (float*) (ws + grab((size_t)XS_ALLOC * 4));
  float*  xtob   = (float*) (ws + grab((size_t)NF_PAD * 16 * 4));
  float*  ytob   = (float*) (ws + grab((size_t)NF_PAD * 16 * 4));
  float*  part2  = (float*) (ws + grab((size_t)SEG_BLOCKS * 4));

  init_tables_kernel<<<(NFRAG + 255) / 256, 256, 0, stream>>>(
      obm, w, frag, obmf, gmask);
  frame_energy_kernel<<<NF_PAD / 8, 256, 0, stream>>>(ref, w, energy);
  max_part_kernel<<<256, 256, 0, stream>>>(energy, partmx);
  max_final_kernel<<<1, 256, 0, stream>>>(partmx, metaf);
  scan_kernel<<<1, 1024, 0, stream>>>(energy, metaf, kept, metai);
  overlap_add_kernel<<<XS_ALLOC / 256, 256, 0, stream>>>(
      ref, inf, w, kept, metai, xsil, ysil);
  dft_band_kernel<<<2048, 64, 0, stream>>>(
      xsil, ysil, frag, obmf, gmask, metai, xtob, ytob);
  seg_corr_kernel<<<SEG_BLOCKS, 256, 0, stream>>>(xtob, ytob, metai, part2);
  finalize_kernel<<<1, 256, 0, stream>>>(part2, metai, out);
}